// PrototypicalNetwork_29351806501536
// MI455X (gfx1250) — compile-verified
//
#include <hip/hip_runtime.h>
#include <hip/hip_bf16.h>
#include <math.h>

// Problem constants (from reference)
#define BATCH 32
#define NSUP  50
#define NQRY  75
#define TLEN  128
#define FDIM  64
#define DDIM  256
#define KWAY  5
#define ROWS  128           // 125 valid rows (50 support + 75 query), padded to 128

typedef __attribute__((ext_vector_type(2))) float v2f;
typedef __attribute__((ext_vector_type(8))) float v8f;

// ---------------------------------------------------------------------------
// Kernel 1: streaming mean-pool over T.  One 256-thread block per sample.
// Reads 32KB/sample via float4 (global_load_b128); ~131MB total -> HBM bound.
// ---------------------------------------------------------------------------
__global__ void pool_kernel(const float4* __restrict__ x,
                            float* __restrict__ pooled,
                            int nPerB, int rowOffset) {
    __shared__ float4 sbuf[256];
    const int tid    = threadIdx.x;
    const int sample = blockIdx.x;                 // b * nPerB + n
    const int b      = sample / nPerB;
    const int n      = sample - b * nPerB;

    // sample = T*F floats = 128 rows * 16 float4
    const float4* src = x + (size_t)sample * (TLEN * (FDIM / 4));

    const int c  = tid & 15;    // float4 column 0..15  (covers F=64)
    const int r0 = tid >> 4;    // 0..15 row group

    float4 acc = make_float4(0.f, 0.f, 0.f, 0.f);
#pragma unroll
    for (int rr = 0; rr < 8; ++rr) {               // 16 groups * 8 = 128 rows
        float4 v = src[(size_t)(r0 + rr * 16) * (FDIM / 4) + c];
        acc.x += v.x; acc.y += v.y; acc.z += v.z; acc.w += v.w;
    }
    sbuf[tid] = acc;
    __syncthreads();

    // reduce across the 16 row-groups sharing column c
    for (int s2 = 8; s2 > 0; s2 >>= 1) {
        if (r0 < s2) {
            float4 o = sbuf[tid + s2 * 16];
            float4 m = sbuf[tid];
            m.x += o.x; m.y += o.y; m.z += o.z; m.w += o.w;
            sbuf[tid] = m;
        }
        __syncthreads();
    }

    if (r0 == 0) {
        const float s = 1.0f / (float)TLEN;
        float4 m = sbuf[c];
        float4* dst = (float4*)(pooled + ((size_t)b * ROWS + rowOffset + n) * FDIM);
        dst[c] = make_float4(m.x * s, m.y * s, m.z * s, m.w * s);
    }
}

// ---------------------------------------------------------------------------
// Kernel 2: Z[b, m, :] = pooled[b, m, :] @ W + bias   via V_WMMA_F32_16X16X4_F32
// One wave32 per 16x16 output tile. 32 batches * 8 M-tiles * 16 N-tiles waves.
// Bias folded into the C accumulator init (bias depends only on N).
// ---------------------------------------------------------------------------
__global__ void encode_wmma_kernel(const float* __restrict__ pooled,
                                   const float* __restrict__ Wm,
                                   const float* __restrict__ bias,
                                   float* __restrict__ Z) {
    const int lane  = threadIdx.x & 31;
    const int wave  = (blockIdx.x * blockDim.x + threadIdx.x) >> 5;
    const int b     = wave >> 7;          // /128 tiles per batch
    const int rem   = wave & 127;
    const int mtile = rem >> 4;           // 0..7
    const int ntile = rem & 15;           // 0..15

    const int row16 = lane & 15;
    const int half  = lane >> 4;          // 0 or 1

    // A: lane holds row M = mtile*16 + row16;  K = k0 + 2*half + {0,1}
    const float* Abase = pooled + ((size_t)b * ROWS + mtile * 16 + row16) * FDIM;
    // B / C / D column for this lane
    const int ncol = ntile * 16 + row16;

    const float bv = bias[ncol];
    v8f acc;
#pragma unroll
    for (int i = 0; i < 8; ++i) acc[i] = bv;

#pragma unroll
    for (int k0 = 0; k0 < FDIM; k0 += 4) {
        const int ka = k0 + 2 * half;
        v2f a;
        a[0] = Abase[ka];
        a[1] = Abase[ka + 1];
        v2f bb;
        bb[0] = Wm[(size_t)ka * DDIM + ncol];
        bb[1] = Wm[(size_t)(ka + 1) * DDIM + ncol];
        // 8 args: (neg_a, A, neg_b, B, c_mod, C, reuse_a, reuse_b)
        acc = __builtin_amdgcn_wmma_f32_16x16x4_f32(
            false, a, false, bb, (short)0, acc, false, false);
    }

    float* Zb = Z + (size_t)b * ROWS * DDIM;
#pragma unroll
    for (int r = 0; r < 8; ++r) {
        const int m = mtile * 16 + r + 8 * half;   // D layout: vgpr r -> M = r (+8 for hi half)
        if (m < NSUP + NQRY)
            Zb[(size_t)m * DDIM + ncol] = acc[r];
    }
}

// ---------------------------------------------------------------------------
// Kernel 3: prototypes via one-hot grouped mean over support rows.
// ---------------------------------------------------------------------------
__global__ void proto_kernel(const float* __restrict__ Z,
                             const int* __restrict__ sy,
                             float* __restrict__ proto) {
    const int idx = blockIdx.x * blockDim.x + threadIdx.x;
    if (idx >= BATCH * KWAY * DDIM) return;
    const int d = idx & (DDIM - 1);
    const int k = (idx >> 8) % KWAY;
    const int b = idx / (KWAY * DDIM);

    const float* Zb = Z + (size_t)b * ROWS * DDIM;
    const int*   yb = sy + b * NSUP;
    float sum = 0.f;
    int   cnt = 0;
    for (int s = 0; s < NSUP; ++s) {
        if (yb[s] == k) { sum += Zb[(size_t)s * DDIM + d]; ++cnt; }
    }
    proto[idx] = sum / (float)cnt;   // idx == (b*KWAY + k)*DDIM + d
}

// ---------------------------------------------------------------------------
// Kernel 4: logits = -||query_z - proto||_2
// ---------------------------------------------------------------------------
__global__ void dist_kernel(const float* __restrict__ Z,
                            const float* __restrict__ proto,
                            float* __restrict__ out) {
    const int idx = blockIdx.x * blockDim.x + threadIdx.x;
    if (idx >= BATCH * NQRY * KWAY) return;
    const int k = idx % KWAY;
    const int q = (idx / KWAY) % NQRY;
    const int b = idx / (NQRY * KWAY);

    const float4* qz = (const float4*)(Z + ((size_t)b * ROWS + NSUP + q) * DDIM);
    const float4* pz = (const float4*)(proto + (size_t)(b * KWAY + k) * DDIM);
    float acc = 0.f;
#pragma unroll 4
    for (int i = 0; i < DDIM / 4; ++i) {
        float4 a = qz[i], p = pz[i];
        float dx = a.x - p.x, dy = a.y - p.y, dz = a.z - p.z, dw = a.w - p.w;
        acc += dx * dx + dy * dy + dz * dz + dw * dw;
    }
    out[idx] = -sqrtf(acc);
}

// ---------------------------------------------------------------------------
extern "C" void kernel_launch(void* const* d_in, const int* in_sizes, int n_in,
                              void* d_out, int out_size, void* d_ws, size_t ws_size,
                              hipStream_t stream) {
    const float* support_x = (const float*)d_in[0];  // [32,50,128,64]
    const int*   support_y = (const int*)  d_in[1];  // [32,50]
    const float* query_x   = (const float*)d_in[2];  // [32,75,128,64]
    const float* W         = (const float*)d_in[3];  // [64,256]
    const float* bias      = (const float*)d_in[4];  // [256]

    char* ws = (char*)d_ws;
    float* pooled = (float*)(ws);                          // 32*128*64*4  = 1 MB
    float* Z      = (float*)(ws + (1u << 20));             // 32*128*256*4 = 4 MB
    float* proto  = (float*)(ws + (5u << 20));             // 32*5*256*4   = 160 KB

    // Streaming mean-pool (the ~131MB HBM-bound part)
    pool_kernel<<<BATCH * NSUP, 256, 0, stream>>>((const float4*)support_x, pooled, NSUP, 0);
    pool_kernel<<<BATCH * NQRY, 256, 0, stream>>>((const float4*)query_x,  pooled, NQRY, NSUP);

    // fp32 WMMA GEMM: 32*8*16 = 4096 wave32 tiles, 8 waves/block
    encode_wmma_kernel<<<(BATCH * 8 * 16) / 8, 256, 0, stream>>>(pooled, W, bias, Z);

    const int nproto = BATCH * KWAY * DDIM;   // 40960
    proto_kernel<<<(nproto + 255) / 256, 256, 0, stream>>>(Z, support_y, proto);

    const int nout = BATCH * NQRY * KWAY;     // 12000
    dist_kernel<<<(nout + 255) / 256, 256, 0, stream>>>(Z, proto, (float*)d_out);
}